// social_stgcn_22247930594050
// MI455X (gfx1250) — compile-verified
//
#include <hip/hip_runtime.h>
#include <cstdint>

typedef float v2f __attribute__((ext_vector_type(2)));
typedef float v8f __attribute__((ext_vector_type(8)));

#define N_NODES 50000
#define N_EDGES 800000
#define F 64
#define SELF_W 2.0f
#define ROW_TILES (N_NODES / 16)   // 3125, exact
#define COL_TILES (F / 16)         // 4
#define N_TILES (ROW_TILES * COL_TILES)

// ---------------------------------------------------------------------------
// 1) deg[i] = 2.0 (self-loop weight, improved=True), agg = 0
// ---------------------------------------------------------------------------
__global__ __launch_bounds__(256) void k_init(float* __restrict__ deg,
                                              float* __restrict__ agg) {
  const long long gid = (long long)blockIdx.x * blockDim.x + threadIdx.x;
  if (gid < N_NODES) deg[gid] = SELF_W;
  const long long total = (long long)N_NODES * F;
  const long long stride = (long long)gridDim.x * blockDim.x;
  for (long long j = gid; j < total; j += stride) agg[j] = 0.0f;
}

// ---------------------------------------------------------------------------
// 2) deg[dst] += 1 per edge
// ---------------------------------------------------------------------------
__global__ __launch_bounds__(256) void k_deg(const long long* __restrict__ dst,
                                             float* __restrict__ deg) {
  const int e = blockIdx.x * blockDim.x + threadIdx.x;
  if (e < N_EDGES) atomicAdd(&deg[(int)dst[e]], 1.0f);
}

// ---------------------------------------------------------------------------
// 3) dinv = deg^-0.5 (deg >= 2 always, but keep the guard like the reference)
// ---------------------------------------------------------------------------
__global__ __launch_bounds__(256) void k_dinv(float* __restrict__ deg) {
  const int i = blockIdx.x * blockDim.x + threadIdx.x;
  if (i < N_NODES) {
    const float d = deg[i];
    deg[i] = (d > 0.0f) ? rsqrtf(d) : 0.0f;
  }
}

// ---------------------------------------------------------------------------
// 4) h = x @ W via V_WMMA_F32_16X16X4_F32.
//    One wave computes one 16x16 tile of h. A (16x4 f32): lanes 0-15 hold
//    K=0,1 in VGPR0,1; lanes 16-31 hold K=2,3. B (4x16): mirrored, N in lanes.
//    C/D (16x16 f32): VGPR r = row r (lanes 0-15) / row 8+r (lanes 16-31).
// ---------------------------------------------------------------------------
__global__ __launch_bounds__(256) void k_gemm(const float* __restrict__ x,
                                              const float* __restrict__ Wm,
                                              float* __restrict__ h) {
  const int wave = (blockIdx.x * blockDim.x + threadIdx.x) >> 5;
  if (wave >= N_TILES) return;               // wave-uniform exit: EXEC all-1s below
  const int lane = threadIdx.x & 31;
  const int rowBase = (wave >> 2) * 16;      // COL_TILES == 4
  const int colBase = (wave & 3) * 16;
  const int half = lane >> 4;                // 0 -> K pair {0,1}, 1 -> {2,3}
  const int m = lane & 15;                   // A: M index, B: N index

  v8f c = {};
#pragma unroll
  for (int kb = 0; kb < 16; ++kb) {
    const int k0 = kb * 4 + half * 2;
    // A fragment: x[rowBase+m][k0], x[rowBase+m][k0+1]  (8-byte aligned)
    v2f a = *(const v2f*)(x + (size_t)(rowBase + m) * F + k0);
    // B fragment: W[k0][colBase+m], W[k0+1][colBase+m]  (W row-major [F_IN,F_OUT])
    v2f b;
    b.x = Wm[(size_t)(k0 + 0) * F + colBase + m];
    b.y = Wm[(size_t)(k0 + 1) * F + colBase + m];
    // 8 args: (neg_a, A, neg_b, B, c_mod, C, reuse_a, reuse_b)
    c = __builtin_amdgcn_wmma_f32_16x16x4_f32(false, a, false, b, (short)0, c,
                                              false, false);
  }
#pragma unroll
  for (int r = 0; r < 8; ++r) {
    const int row = rowBase + half * 8 + r;
    h[(size_t)row * F + colBase + m] = c[r];
  }
}

// ---------------------------------------------------------------------------
// 5) One wave per edge: agg[dst] += dinv[src]*dinv[dst] * h[src]
//    Each lane owns a float2 of the 64 features -> 2 f32 atomics per lane.
// ---------------------------------------------------------------------------
__global__ __launch_bounds__(256) void k_scatter(const long long* __restrict__ src,
                                                 const long long* __restrict__ dst,
                                                 const float* __restrict__ dinv,
                                                 const float* __restrict__ h,
                                                 float* __restrict__ agg) {
  const int e = (int)(((long long)blockIdx.x * blockDim.x + threadIdx.x) >> 5);
  if (e >= N_EDGES) return;
  const int lane = threadIdx.x & 31;
  const int s = (int)src[e];
  const int d = (int)dst[e];
  const float nrm = dinv[s] * dinv[d];
  const int f0 = lane * 2;
  v2f hv = *(const v2f*)(h + (size_t)s * F + f0);
  atomicAdd(&agg[(size_t)d * F + f0 + 0], nrm * hv.x);
  atomicAdd(&agg[(size_t)d * F + f0 + 1], nrm * hv.y);
}

// ---------------------------------------------------------------------------
// 6) One wave per node: + self-loop + bias, ReLU, log_softmax over 64 feats.
//    Lane l owns features 2l, 2l+1; wave32 shfl_xor tree reductions.
// ---------------------------------------------------------------------------
__global__ __launch_bounds__(256) void k_final(const float* __restrict__ agg,
                                               const float* __restrict__ h,
                                               const float* __restrict__ dinv,
                                               const float* __restrict__ bias,
                                               float* __restrict__ out) {
  const int node = (int)(((long long)blockIdx.x * blockDim.x + threadIdx.x) >> 5);
  if (node >= N_NODES) return;
  const int lane = threadIdx.x & 31;
  const float di = dinv[node];
  const float selfn = SELF_W * di * di;
  const int f0 = lane * 2;
  const size_t base = (size_t)node * F + f0;
  float v0 = agg[base + 0] + selfn * h[base + 0] + bias[f0 + 0];
  float v1 = agg[base + 1] + selfn * h[base + 1] + bias[f0 + 1];
  v0 = fmaxf(v0, 0.0f);
  v1 = fmaxf(v1, 0.0f);

  float mx = fmaxf(v0, v1);
#pragma unroll
  for (int off = 16; off > 0; off >>= 1) mx = fmaxf(mx, __shfl_xor(mx, off, 32));
  float s = expf(v0 - mx) + expf(v1 - mx);
#pragma unroll
  for (int off = 16; off > 0; off >>= 1) s += __shfl_xor(s, off, 32);
  const float lse = mx + logf(s);
  out[base + 0] = v0 - lse;
  out[base + 1] = v1 - lse;
}

// ---------------------------------------------------------------------------
extern "C" void kernel_launch(void* const* d_in, const int* in_sizes, int n_in,
                              void* d_out, int out_size, void* d_ws, size_t ws_size,
                              hipStream_t stream) {
  const float* x = (const float*)d_in[0];
  const long long* ei = (const long long*)d_in[1];   // int64 [2, E]
  const float* Wm = (const float*)d_in[2];
  const float* bias = (const float*)d_in[3];
  float* out = (float*)d_out;

  const long long* src = ei;
  const long long* dst = ei + N_EDGES;

  // Workspace layout (floats): [deg/dinv : N][h : N*F][agg : N*F] = ~25.8 MB
  float* ws = (float*)d_ws;
  float* deg = ws;                       // becomes dinv after k_dinv
  float* h = ws + N_NODES;               // 200000 B offset -> 16B aligned
  float* agg = h + (size_t)N_NODES * F;

  (void)in_sizes; (void)n_in; (void)out_size; (void)ws_size;

  // 1) init: deg = 2.0, agg = 0            (N*F / 256 = 12500 blocks)
  k_init<<<12500, 256, 0, stream>>>(deg, agg);
  // 2) degree accumulation                 (E / 256 = 3125 blocks)
  k_deg<<<3125, 256, 0, stream>>>(dst, deg);
  // 3) dinv = deg^-0.5
  k_dinv<<<(N_NODES + 255) / 256, 256, 0, stream>>>(deg);
  // 4) h = x @ W  (12500 tiles, 8 waves/block)
  k_gemm<<<(N_TILES + 7) / 8, 256, 0, stream>>>(x, Wm, h);
  // 5) edge scatter (one wave per edge: E*32 threads)
  k_scatter<<<(int)(((long long)N_EDGES * 32 + 255) / 256), 256, 0, stream>>>(
      src, dst, deg, h, agg);
  // 6) finalize (one wave per node: N*32 threads)
  k_final<<<(int)(((long long)N_NODES * 32 + 255) / 256), 256, 0, stream>>>(
      agg, h, deg, bias, out);
}